// Featurizer_72808285602207
// MI455X (gfx1250) — compile-verified
//
#include <hip/hip_runtime.h>
#include <math.h>

typedef float v2f __attribute__((ext_vector_type(2)));
typedef float v8f __attribute__((ext_vector_type(8)));

#define HD 128
#define FEPS 1e-6f

// ---------- small vec helpers ----------
__device__ __forceinline__ float3 vsub3(float3 a, float3 b) { return make_float3(a.x-b.x, a.y-b.y, a.z-b.z); }
__device__ __forceinline__ float  vdot3(float3 a, float3 b) { return a.x*b.x + a.y*b.y + a.z*b.z; }
__device__ __forceinline__ float3 vcross3(float3 a, float3 b) {
    return make_float3(a.y*b.z - a.z*b.y, a.z*b.x - a.x*b.z, a.x*b.y - a.y*b.x);
}
__device__ __forceinline__ float3 vnorm3(float3 a) {
    float n = fmaxf(sqrtf(vdot3(a,a)), 1e-12f);
    float inv = 1.0f / n;
    return make_float3(a.x*inv, a.y*inv, a.z*inv);
}
__device__ __forceinline__ float sgnf(float x) { return (x > 0.f) ? 1.f : ((x < 0.f) ? -1.f : 0.f); }

// ds_swizzle group-of-32 xor-butterfly add: pattern = (xor<<10)|(or<<5)|and_mask
// xor masks 1/2/4/8 stay inside each 16-lane half (never touch lane bit 4).
template<int PAT>
__device__ __forceinline__ float swz_add(float v) {
    return v + __int_as_float(__builtin_amdgcn_ds_swizzle(__float_as_int(v), PAT));
}
__device__ __forceinline__ float half16_sum(float v) {
    v = swz_add<0x041f>(v);   // xor 1
    v = swz_add<0x081f>(v);   // xor 2
    v = swz_add<0x101f>(v);   // xor 4
    v = swz_add<0x201f>(v);   // xor 8
    return v;                 // every lane in its 16-lane half holds the half-sum
}

// =====================================================================
// Kernel 1: per-node geometry -> Q frames [N,9], node feats [N,4], invq[N]
// =====================================================================
__global__ void geom_kernel(const float* __restrict__ X,
                            const int* __restrict__ batch,
                            const int* __restrict__ chain,
                            float* __restrict__ Qout,
                            float* __restrict__ nfeat,
                            unsigned char* __restrict__ invq,
                            int N)
{
    int i = blockIdx.x * blockDim.x + threadIdx.x;
    if (i >= N) return;

    auto change = [&](int j) -> bool {   // valid for j in [0, N-2]
        return (batch[j] != batch[j+1]) || (chain[j] != chain[j+1]);
    };
    bool m1  = (i >= 1)     && change(i-1);
    bool m0  = (i <= N-2)   && change(i);
    bool mm1 = (i <= N-3)   && change(i+1);
    bool iq  = m1 || m0 || (i == 0) || (i == N-1);
    bool idh = iq || mm1 || (i == N-2);

    auto loadX = [&](int j) -> float3 {
        j = min(max(j, 0), N-1);
        return make_float3(X[3*j+0], X[3*j+1], X[3*j+2]);
    };
    float3 Xm1 = loadX(i-1), X0 = loadX(i), Xp1 = loadX(i+1), Xp2 = loadX(i+2);
    float3 Um1 = vnorm3(vsub3(X0,  Xm1));   // U[i-1]
    float3 U0  = vnorm3(vsub3(Xp1, X0));    // U[i]
    float3 Up1 = vnorm3(vsub3(Xp2, Xp1));   // U[i+1]

    // dihedral
    float sin_d = 0.f, cos_d = 0.f;
    if (!idh) {
        float3 c1 = vnorm3(vcross3(Um1, U0));
        float3 c2 = vnorm3(vcross3(U0, Up1));
        float cd = fminf(fmaxf(vdot3(c1, c2), -1.f + FEPS), 1.f - FEPS);
        float d = acosf(cd) * sgnf(vdot3(c2, Um1));
        sin_d = sinf(d);
        cos_d = cosf(d);
    }

    // bond angle + local frame
    float sin_a = 0.f, cos_a = 0.f;
    float q[9];
    #pragma unroll
    for (int k = 0; k < 9; ++k) q[k] = 0.f;
    if (!iq) {
        float3 d0 = vnorm3(vsub3(Xm1, X0));
        float3 d1 = vnorm3(vsub3(Xp1, X0));
        cos_a = vdot3(d0, d1);
        sin_a = sqrtf(1.f - cos_a*cos_a + FEPS);

        float3 b  = vnorm3(vsub3(Um1, U0));
        float3 n  = vnorm3(vcross3(Um1, U0));
        float3 bn = vcross3(b, n);
        // Q[r][c]: columns are b, n, b x n  (row-major 9 floats)
        q[0] = b.x; q[1] = n.x; q[2] = bn.x;
        q[3] = b.y; q[4] = n.y; q[5] = bn.y;
        q[6] = b.z; q[7] = n.z; q[8] = bn.z;
    }

    long long o = (long long)i * 9;
    #pragma unroll
    for (int k = 0; k < 9; ++k) Qout[o + k] = q[k];
    long long f = (long long)i * 4;
    nfeat[f+0] = sin_d; nfeat[f+1] = cos_d; nfeat[f+2] = sin_a; nfeat[f+3] = cos_a;
    invq[i] = iq ? 1 : 0;
}

// =====================================================================
// Shared epilogue: 16 rows x 128 ch GEMM via V_WMMA_F32_16X16X4_F32 + LayerNorm
// Block = 256 threads = 8 waves; wave w owns channels [16w, 16w+16).
// =====================================================================
template<int KPAD>
__device__ __forceinline__ void gemm_ln_16rows(
    const float* feat_sh,              // LDS [16][KPAD], cols >= K zeroed
    int K,
    const float* __restrict__ W,       // [K][128] row-major
    const float* __restrict__ bias,    // [128]
    const float* __restrict__ g,       // [128]
    const float* __restrict__ be,      // [128]
    float* __restrict__ out,           // [rows][128]
    long long row0, long long rows,
    float* red_sum, float* red_sq, float* mu_sh, float* rs_sh)
{
    const int tid  = threadIdx.x;
    const int w    = tid >> 5;
    const int lane = tid & 31;
    const int half = lane >> 4;        // 0: lanes 0-15, 1: lanes 16-31
    const int l16  = lane & 15;
    const int n    = w * 16 + l16;     // output channel

    v8f c = {0.f,0.f,0.f,0.f,0.f,0.f,0.f,0.f};
    #pragma unroll
    for (int k0 = 0; k0 < KPAD; k0 += 4) {
        const int ka = k0 + (half ? 2 : 0);
        v2f a, b;
        // A: 16x4 f32 layout — lane m holds K=ka, ka+1
        a.x = feat_sh[l16 * KPAD + ka];
        a.y = feat_sh[l16 * KPAD + ka + 1];
        // B: 4x16 f32 layout — lane n holds rows K=ka, ka+1 (guard K pad)
        b.x = (ka     < K) ? W[(long long)ka * HD + n]     : 0.f;
        b.y = (ka + 1 < K) ? W[(long long)(ka+1) * HD + n] : 0.f;
        c = __builtin_amdgcn_wmma_f32_16x16x4_f32(false, a, false, b, (short)0, c, false, false);
    }

    // bias + per-(wave,row) partial sums over 16 channels (ds_swizzle butterflies)
    const float bn = bias[n];
    float vals[8];
    #pragma unroll
    for (int r = 0; r < 8; ++r) {
        const int m = r + half * 8;    // row within tile
        float v = c[r] + bn;
        vals[r] = v;
        float s = half16_sum(v);
        float q = half16_sum(v * v);
        if (l16 == 0) { red_sum[w * 16 + m] = s; red_sq[w * 16 + m] = q; }
    }
    __syncthreads();
    if (tid < 16) {
        float s = 0.f, q = 0.f;
        #pragma unroll
        for (int ww = 0; ww < 8; ++ww) { s += red_sum[ww*16 + tid]; q += red_sq[ww*16 + tid]; }
        float mu  = s * (1.0f / 128.0f);
        float var = q * (1.0f / 128.0f) - mu * mu;
        mu_sh[tid] = mu;
        rs_sh[tid] = rsqrtf(var + 1e-5f);
    }
    __syncthreads();

    const float gg = g[n], bb = be[n];
    #pragma unroll
    for (int r = 0; r < 8; ++r) {
        const int m = r + half * 8;
        const long long row = row0 + m;
        if (row < rows) {
            float o = (vals[r] - mu_sh[m]) * rs_sh[m] * gg + bb;
            __builtin_nontemporal_store(o, &out[row * HD + n]);   // streaming: bypass L2 retention
        }
    }
}

// =====================================================================
// Kernel 2: edge features (quat, rbf, direction) + WMMA GEMM + LN
// =====================================================================
__global__ void __launch_bounds__(256)
edge_kernel(const float* __restrict__ X,
            const int* __restrict__ ei,
            const float* __restrict__ Q,
            const unsigned char* __restrict__ invq,
            const float* __restrict__ W,
            const float* __restrict__ bias,
            const float* __restrict__ g,
            const float* __restrict__ be,
            float* __restrict__ out,
            long long E, int N)
{
    __shared__ float feat_sh[16 * 24];
    __shared__ float dist_sh[16];
    __shared__ float red_sum[8 * 16], red_sq[8 * 16], mu_sh[16], rs_sh[16];

    const int tid = threadIdx.x;
    const long long e0 = (long long)blockIdx.x * 16;

    if (tid < 16) {
        const long long e = e0 + tid;
        float f[24];
        #pragma unroll
        for (int k = 0; k < 24; ++k) f[k] = 0.f;
        float dist = 0.f;
        if (e < E) {
            const int src = ei[e];
            const int tgt = ei[E + e];
            const float* Qs = Q + (long long)src * 9;
            const float* Qt = Q + (long long)tgt * 9;
            const bool einv = (invq[src] | invq[tgt]) != 0;

            // R = Qt^T * Qs : R[i][k] = sum_j Qt[j][i] * Qs[j][k]
            float R[3][3];
            #pragma unroll
            for (int ii = 0; ii < 3; ++ii)
                #pragma unroll
                for (int kk = 0; kk < 3; ++kk)
                    R[ii][kk] = Qt[0*3+ii]*Qs[0*3+kk] + Qt[1*3+ii]*Qs[1*3+kk] + Qt[2*3+ii]*Qs[2*3+kk];

            const float Rxx = R[0][0], Ryy = R[1][1], Rzz = R[2][2];
            const float mx = 0.5f * sqrtf(fabsf(1.f + Rxx - Ryy - Rzz));
            const float my = 0.5f * sqrtf(fabsf(1.f - Rxx + Ryy - Rzz));
            const float mz = 0.5f * sqrtf(fabsf(1.f - Rxx - Ryy + Rzz));
            const float qx = sgnf(R[2][1] - R[1][2]) * mx;
            const float qy = sgnf(R[0][2] - R[2][0]) * my;
            const float qz = sgnf(R[1][0] - R[0][1]) * mz;
            const float qw = sqrtf(fmaxf(1.f + Rxx + Ryy + Rzz, 0.f)) * 0.5f;
            const float qinv = 1.f / fmaxf(sqrtf(qx*qx + qy*qy + qz*qz + qw*qw), 1e-12f);
            if (!einv) { f[0] = qx*qinv; f[1] = qy*qinv; f[2] = qz*qinv; f[3] = qw*qinv; }

            const float3 Xs = make_float3(X[3*src+0], X[3*src+1], X[3*src+2]);
            const float3 Xt = make_float3(X[3*tgt+0], X[3*tgt+1], X[3*tgt+2]);
            const float3 d  = vsub3(Xs, Xt);
            dist = sqrtf(vdot3(d, d) + FEPS);
            const float3 nd = vnorm3(d);
            // direct = Qt^T * nd  (Qt is zero when invq[tgt] -> matches reference mask)
            f[20] = Qt[0]*nd.x + Qt[3]*nd.y + Qt[6]*nd.z;
            f[21] = Qt[1]*nd.x + Qt[4]*nd.y + Qt[7]*nd.z;
            f[22] = Qt[2]*nd.x + Qt[5]*nd.y + Qt[8]*nd.z;
        }
        dist_sh[tid] = dist;
        #pragma unroll
        for (int k = 0; k < 24; ++k) feat_sh[tid * 24 + k] = f[k];
    }
    __syncthreads();

    // RBF: one thread per (edge, rbf-bin); mu_k = 20k/15, sigma = 1.25
    {
        const int el = tid >> 4, k = tid & 15;
        const float dist = dist_sh[el];
        const float t = (dist - (20.0f / 15.0f) * (float)k) * (1.0f / 1.25f);
        feat_sh[el * 24 + 4 + k] = expf(-t * t);
    }
    __syncthreads();

    gemm_ln_16rows<24>(feat_sh, 23, W, bias, g, be, out, e0, E,
                       red_sum, red_sq, mu_sh, rs_sh);
    (void)N;
}

// =====================================================================
// Kernel 3: node GEMM (K=4, single WMMA) + LN
// =====================================================================
__global__ void __launch_bounds__(256)
node_out_kernel(const float* __restrict__ nfeat,
                const float* __restrict__ W,
                const float* __restrict__ bias,
                const float* __restrict__ g,
                const float* __restrict__ be,
                float* __restrict__ out,
                long long Nn)
{
    __shared__ float feat_sh[16 * 4];
    __shared__ float red_sum[8 * 16], red_sq[8 * 16], mu_sh[16], rs_sh[16];

    const int tid = threadIdx.x;
    const long long r0 = (long long)blockIdx.x * 16;
    if (tid < 64) {
        const int m = tid >> 2, k = tid & 3;
        const long long row = r0 + m;
        feat_sh[m * 4 + k] = (row < Nn) ? nfeat[row * 4 + k] : 0.f;
    }
    __syncthreads();

    gemm_ln_16rows<4>(feat_sh, 4, W, bias, g, be, out, r0, Nn,
                      red_sum, red_sq, mu_sh, rs_sh);
}

// =====================================================================
// Host launch
// =====================================================================
extern "C" void kernel_launch(void* const* d_in, const int* in_sizes, int n_in,
                              void* d_out, int out_size, void* d_ws, size_t ws_size,
                              hipStream_t stream)
{
    const float* X      = (const float*)d_in[0];
    const int*   batch  = (const int*)d_in[1];
    const int*   chain  = (const int*)d_in[2];
    const int*   ei     = (const int*)d_in[3];
    const float* W_node = (const float*)d_in[4];
    const float* b_node = (const float*)d_in[5];
    const float* W_edge = (const float*)d_in[6];
    const float* b_edge = (const float*)d_in[7];
    const float* g_node = (const float*)d_in[8];
    const float* be_node= (const float*)d_in[9];
    const float* g_edge = (const float*)d_in[10];
    const float* be_edge= (const float*)d_in[11];
    (void)n_in; (void)out_size; (void)ws_size;

    const int N = in_sizes[0] / 3;
    const long long E = (long long)(in_sizes[3] / 2);

    // workspace layout
    float* Qws     = (float*)d_ws;                       // N*9 floats
    float* nfeat   = Qws + (size_t)N * 9;                // N*4 floats
    unsigned char* invq = (unsigned char*)(nfeat + (size_t)N * 4); // N bytes

    float* out_hV = (float*)d_out;                       // [N,128]
    float* out_hE = out_hV + (long long)N * HD;          // [E,128]

    // 1. geometry
    {
        dim3 blk(256);
        dim3 grd((N + 255) / 256);
        geom_kernel<<<grd, blk, 0, stream>>>(X, batch, chain, Qws, nfeat, invq, N);
    }
    // 2. node output (WMMA K=4 + LN)
    {
        dim3 blk(256);
        dim3 grd((unsigned)((N + 15) / 16));
        node_out_kernel<<<grd, blk, 0, stream>>>(nfeat, W_node, b_node, g_node, be_node,
                                                 out_hV, (long long)N);
    }
    // 3. edge output (features + WMMA K=24 + LN)
    {
        dim3 blk(256);
        dim3 grd((unsigned)((E + 15) / 16));
        edge_kernel<<<grd, blk, 0, stream>>>(X, ei, Qws, invq, W_edge, b_edge,
                                             g_edge, be_edge, out_hE, E, N);
    }
}